// N3Block_7335804142147
// MI455X (gfx1250) — compile-verified
//
#include <hip/hip_runtime.h>
#include <math.h>

// ---------------- problem constants ----------------
#define IMG    320
#define BATCH  2
#define FD     64
#define OC     8
#define PP     10      // patch size
#define SS     5       // stride
#define FH     63      // (320-10)/5+1
#define NPATCH 3969    // 63*63
#define NPAD   3984    // 249*16, WMMA-padded
#define NT     249     // NPAD/16
#define CDIM   800     // OC*PP*PP
#define KNN    7

typedef float v2f __attribute__((ext_vector_type(2)));
typedef float v8f __attribute__((ext_vector_type(8)));

// ---------------------------------------------------------------------------
// Fused 3x3 conv (+bias, optional BN*gain+beta and ReLU), pad=1.
// grid: (400 spatial 16x16 tiles, Cout, B); block: 256
// ---------------------------------------------------------------------------
__global__ __launch_bounds__(256)
void conv3x3_kernel(const float* __restrict__ in, const float* __restrict__ wgt,
                    const float* __restrict__ bias, const float* __restrict__ gain,
                    const float* __restrict__ beta, float* __restrict__ out,
                    int Cin, int Cout, int fuseBnRelu) {
  const int tile = blockIdx.x;
  const int co   = blockIdx.y;
  const int b    = blockIdx.z;
  const int th0  = (tile / 20) * 16;
  const int tw0  = (tile % 20) * 16;
  const int ty   = threadIdx.x >> 4;
  const int tx   = threadIdx.x & 15;
  const int h    = th0 + ty, w = tw0 + tx;

  __shared__ float sIn[18 * 18];
  __shared__ float sW[FD * 9];

  for (int i = threadIdx.x; i < Cin * 9; i += 256)
    sW[i] = wgt[(size_t)co * Cin * 9 + i];

  float acc = 0.0f;
  for (int ci = 0; ci < Cin; ++ci) {
    __syncthreads();                       // protects prior reads of sIn (and sW preload)
    const float* ip = in + (size_t)(b * Cin + ci) * IMG * IMG;
    for (int e = threadIdx.x; e < 18 * 18; e += 256) {
      int ih = th0 + (e / 18) - 1;
      int iw = tw0 + (e % 18) - 1;
      sIn[e] = (ih >= 0 && ih < IMG && iw >= 0 && iw < IMG) ? ip[ih * IMG + iw] : 0.0f;
    }
    __syncthreads();
    const float* wp = &sW[ci * 9];
    const float* c0 = &sIn[ty * 18 + tx];
    acc = fmaf(c0[0],  wp[0], acc); acc = fmaf(c0[1],  wp[1], acc); acc = fmaf(c0[2],  wp[2], acc);
    acc = fmaf(c0[18], wp[3], acc); acc = fmaf(c0[19], wp[4], acc); acc = fmaf(c0[20], wp[5], acc);
    acc = fmaf(c0[36], wp[6], acc); acc = fmaf(c0[37], wp[7], acc); acc = fmaf(c0[38], wp[8], acc);
  }
  float y = acc + bias[co];
  if (fuseBnRelu) {
    const float inv = 0.9999950000374997f;  // 1/sqrt(1+1e-5)
    y = fmaxf(fmaf(y, gain[co] * inv, beta[co]), 0.0f);
  }
  out[((size_t)(b * Cout + co) * IMG + h) * IMG + w] = y;
}

// ---------------------------------------------------------------------------
// Patch extraction -> E2col [B][NPAD][CDIM], zero-padded rows beyond NPATCH.
// ---------------------------------------------------------------------------
__global__ __launch_bounds__(256)
void e2col_kernel(const float* __restrict__ e3, float* __restrict__ e2col) {
  size_t t = (size_t)blockIdx.x * 256 + threadIdx.x;
  const size_t total = (size_t)BATCH * NPAD * CDIM;
  if (t >= total) return;
  int c = (int)(t % CDIM);
  int n = (int)((t / CDIM) % NPAD);
  int b = (int)(t / ((size_t)CDIM * NPAD));
  float v = 0.0f;
  if (n < NPATCH) {
    int i = n / FH, j = n % FH;
    int ch = c / 100, r = (c / 10) % 10, q = c % 10;
    v = e3[((size_t)(b * OC + ch) * IMG + (i * SS + r)) * IMG + (j * SS + q)];
  }
  e2col[t] = v;
}

// ---------------------------------------------------------------------------
// Row squared-norms. One wave per row, wave32 shuffle reduction.
// ---------------------------------------------------------------------------
__global__ __launch_bounds__(256)
void sq_kernel(const float* __restrict__ e2col, float* __restrict__ sq) {
  int row  = blockIdx.x * 8 + (threadIdx.x >> 5);
  int lane = threadIdx.x & 31;
  if (row >= BATCH * NPAD) return;
  const float* p = e2col + (size_t)row * CDIM;
  float s = 0.0f;
  for (int c = lane; c < CDIM; c += 32) { float v = p[c]; s = fmaf(v, v, s); }
  #pragma unroll
  for (int off = 16; off > 0; off >>= 1) s += __shfl_xor(s, off, 32);
  if (lane == 0) sq[row] = s;
}

// ---------------------------------------------------------------------------
// Fused Gram (f32 WMMA 16x16x4) + distance + per-row top-7 selection.
// grid: (NT row tiles, BATCH); block: 128 (4 waves).
// A panel (16 rows x 800) staged in LDS, shared by all waves.
// Each wave sweeps interleaved 16-col tiles; 200 v_wmma per tile.
// ---------------------------------------------------------------------------
__global__ __launch_bounds__(128)
void gram_topk_kernel(const float* __restrict__ e2, const float* __restrict__ sq,
                      int* __restrict__ idxOut) {
  const int b       = blockIdx.y;
  const int rowBase = blockIdx.x * 16;
  const int tid     = threadIdx.x;
  const int wave    = tid >> 5;
  const int lane    = tid & 31;
  const int m16     = lane & 15;
  const int kb      = (lane >> 4) << 1;   // 0 or 2: K base for this half-wave

  __shared__ __align__(16) float ldsA[16 * CDIM];   // 51.2 KB row panel
  __shared__ float ldsG[4][256];                    // per-wave 16x16 Gram scratch
  __shared__ float mVal[4][16][KNN];
  __shared__ int   mId [4][16][KNN];

  {
    const float* src = e2 + ((size_t)b * NPAD + rowBase) * CDIM;  // 16 contiguous rows
    for (int i = tid; i < 16 * CDIM; i += 128) ldsA[i] = src[i];
  }
  __syncthreads();

  const float sqR  = sq[b * NPAD + rowBase + m16];
  const int   gRow = rowBase + m16;      // global row tracked by lanes 0..15

  float topV[KNN]; int topI[KNN];
  #pragma unroll
  for (int k = 0; k < KNN; ++k) { topV[k] = -INFINITY; topI[k] = 0x7fffffff; }

  for (int ct = wave; ct < NT; ct += 4) {
    const int colBase = ct * 16;
    v8f acc = {};
    const float* aP = &ldsA[m16 * CDIM + kb];
    const float* bP = e2 + ((size_t)b * NPAD + colBase + m16) * CDIM + kb;
    #pragma unroll 4
    for (int c0 = 0; c0 < CDIM; c0 += 4) {
      v2f av = *(const v2f*)(aP + c0);   // A[m][c0+kb], A[m][c0+kb+1]   (LDS)
      v2f bv = *(const v2f*)(bP + c0);   // B[n][c0+kb], B[n][c0+kb+1]   (global/L2)
      acc = __builtin_amdgcn_wmma_f32_16x16x4_f32(
          /*neg_a=*/false, av, /*neg_b=*/false, bv,
          /*c_mod=*/(short)0, acc, /*reuse_a=*/false, /*reuse_b=*/false);
    }
    // Transpose the accumulator through per-wave LDS:
    // lane holds column n=m16, rows v + 8*(lane>>4), v=0..7
    {
      float* g = ldsG[wave];
      const int mOff = (lane >> 4) << 3;
      #pragma unroll
      for (int v = 0; v < 8; ++v) g[(mOff + v) * 16 + m16] = acc[v];
    }
    __asm volatile("s_wait_dscnt 0" ::: "memory");  // intra-wave LDS RAW
    if (lane < 16) {                                 // lane = local row
      const float* g = &ldsG[wave][lane * 16];
      #pragma unroll 1
      for (int nl = 0; nl < 16; ++nl) {
        int n = colBase + nl;
        float neg;
        if (n == gRow || n >= NPATCH) {
          neg = -INFINITY;
        } else {
          float d2 = sqR + sq[b * NPAD + n] - 2.0f * g[nl];
          neg = -sqrtf(fmaxf(d2, 0.0f));
        }
        if (neg > topV[KNN - 1]) {                   // strict: ties keep lower index
          int k = KNN - 1;
          while (k > 0 && topV[k - 1] < neg) {
            topV[k] = topV[k - 1]; topI[k] = topI[k - 1]; --k;
          }
          topV[k] = neg; topI[k] = n;
        }
      }
    }
  }

  if (lane < 16) {
    #pragma unroll
    for (int k = 0; k < KNN; ++k) { mVal[wave][lane][k] = topV[k]; mId[wave][lane][k] = topI[k]; }
  }
  __syncthreads();

  if (tid < 16) {
    const int gr = rowBase + tid;
    if (gr < NPATCH) {
      float cv[4 * KNN]; int ci[4 * KNN];
      #pragma unroll
      for (int wv = 0; wv < 4; ++wv)
        #pragma unroll
        for (int k = 0; k < KNN; ++k) {
          cv[wv * KNN + k] = mVal[wv][tid][k];
          ci[wv * KNN + k] = mId [wv][tid][k];
        }
      int* dst = idxOut + ((size_t)b * NPATCH + gr) * KNN;
      for (int k = 0; k < KNN; ++k) {
        int best = 0;
        for (int c = 1; c < 4 * KNN; ++c)
          if (cv[c] > cv[best] || (cv[c] == cv[best] && ci[c] < ci[best])) best = c;
        dst[k] = ci[best];
        cv[best] = -INFINITY; ci[best] = 0x7fffffff;
      }
    }
  }
}

// ---------------------------------------------------------------------------
// Gather-based overlap-add fold: each output pixel sums its <=4 patch cells.
// cf = k*OC + ch ; k==0 -> self patch, k>=1 -> neighbor idx[b][n][k-1].
// ---------------------------------------------------------------------------
__global__ __launch_bounds__(256)
void fold_kernel(const float* __restrict__ e2col, const int* __restrict__ idx,
                 float* __restrict__ out) {
  size_t t = (size_t)blockIdx.x * 256 + threadIdx.x;
  const size_t total = (size_t)BATCH * FD * IMG * IMG;
  if (t >= total) return;
  int w  = (int)(t % IMG);
  int h  = (int)((t / IMG) % IMG);
  int cf = (int)((t / (IMG * IMG)) % FD);
  int b  = (int)(t / ((size_t)IMG * IMG * FD));
  int k  = cf >> 3, ch = cf & 7;

  float acc = 0.0f;
  int i1 = h / SS, j1 = w / SS;
  for (int i = i1 - 1; i <= i1; ++i) {
    if (i < 0 || i >= FH) continue;
    int pi = h - i * SS;
    if (pi >= PP) continue;
    for (int j = j1 - 1; j <= j1; ++j) {
      if (j < 0 || j >= FH) continue;
      int pj = w - j * SS;
      if (pj >= PP) continue;
      int n   = i * FH + j;
      int src = (k == 0) ? n : idx[((size_t)b * NPATCH + n) * KNN + (k - 1)];
      acc += e2col[((size_t)b * NPAD + src) * CDIM + (ch * 100 + pi * 10 + pj)];
    }
  }
  out[t] = acc;
}

// ---------------------------------------------------------------------------
extern "C" void kernel_launch(void* const* d_in, const int* in_sizes, int n_in,
                              void* d_out, int out_size, void* d_ws, size_t ws_size,
                              hipStream_t stream) {
  (void)in_sizes; (void)n_in; (void)out_size; (void)ws_size;
  const float* x   = (const float*)d_in[0];
  const float* w1  = (const float*)d_in[1];
  const float* b1  = (const float*)d_in[2];
  const float* g1  = (const float*)d_in[3];
  const float* be1 = (const float*)d_in[4];
  const float* w2  = (const float*)d_in[5];
  const float* b2  = (const float*)d_in[6];
  const float* g2  = (const float*)d_in[7];
  const float* be2 = (const float*)d_in[8];
  const float* w3  = (const float*)d_in[9];
  const float* b3  = (const float*)d_in[10];
  float* out = (float*)d_out;

  // workspace layout (all 1 KB aligned)
  constexpr size_t SZ_BUF   = (size_t)BATCH * FD * IMG * IMG * 4;   // 52,428,800
  constexpr size_t SZ_E2COL = (size_t)BATCH * NPAD * CDIM * 4;      // 25,497,600
  constexpr size_t SZ_SQ    = ((size_t)BATCH * NPAD * 4 + 1023) & ~(size_t)1023;
  char* ws = (char*)d_ws;
  float* bufA  = (float*)ws;                                   // E1, then E3
  float* bufB  = (float*)(ws + SZ_BUF);                        // E2
  float* e2col = (float*)(ws + 2 * SZ_BUF);
  float* sqv   = (float*)(ws + 2 * SZ_BUF + SZ_E2COL);
  int*   idxb  = (int*)  (ws + 2 * SZ_BUF + SZ_E2COL + SZ_SQ);

  // conv1: x(3ch) -> bufA(64ch), BN+ReLU
  dim3 cg1(400, FD, BATCH);
  conv3x3_kernel<<<cg1, 256, 0, stream>>>(x, w1, b1, g1, be1, bufA, 3, FD, 1);
  // conv2: bufA -> bufB, BN+ReLU
  conv3x3_kernel<<<cg1, 256, 0, stream>>>(bufA, w2, b2, g2, be2, bufB, FD, FD, 1);
  // conv3: bufB -> bufA(8ch), bias only
  dim3 cg3(400, OC, BATCH);
  conv3x3_kernel<<<cg3, 256, 0, stream>>>(bufB, w3, b3, nullptr, nullptr, bufA, FD, OC, 0);

  // patches -> E2col (zero-padded rows)
  {
    size_t tot = (size_t)BATCH * NPAD * CDIM;
    e2col_kernel<<<(unsigned)((tot + 255) / 256), 256, 0, stream>>>(bufA, e2col);
  }
  // squared norms
  sq_kernel<<<(BATCH * NPAD + 7) / 8, 256, 0, stream>>>(e2col, sqv);
  // fused WMMA Gram + kNN selection
  {
    dim3 gg(NT, BATCH);
    gram_topk_kernel<<<gg, 128, 0, stream>>>(e2col, sqv, idxb);
  }
  // gather + overlap-add fold
  {
    size_t tot = (size_t)BATCH * FD * IMG * IMG;
    fold_kernel<<<(unsigned)((tot + 255) / 256), 256, 0, stream>>>(e2col, idxb, out);
  }
}